// CliffordAlgebra_18537078849843
// MI455X (gfx1250) — compile-verified
//
#include <hip/hip_runtime.h>
#include <cstdint>

typedef float v2f __attribute__((ext_vector_type(2)));

// ---------------------------------------------------------------------------
// Compile-time Cayley sign tables for Cl(3,1,0), D=16.
// e_a * e_b = sign(a,b) * e_{a XOR b}; metric = [+1,+1,+1,-1].
// Faithful constexpr port of the Python _blade_sign.
// ---------------------------------------------------------------------------
constexpr float metric_of(int b) { return (b == 3) ? -1.0f : 1.0f; }

constexpr float blade_sign(int a_idx, int b_idx) {
    int bits[8] = {0, 0, 0, 0, 0, 0, 0, 0};
    int n = 0;
    for (int i = 0; i < 4; ++i)
        if ((a_idx >> i) & 1) bits[n++] = i;
    float sign = 1.0f;
    for (int b = 0; b < 4; ++b) {
        if (!((b_idx >> b) & 1)) continue;
        int pos = n, swaps = 0;
        for (int i = n - 1; i >= 0; --i) {
            if (bits[i] < b) break;
            if (bits[i] == b) {
                sign *= metric_of(b);
                for (int k = i; k < n - 1; ++k) bits[k] = bits[k + 1];
                --n; pos = -1; break;
            }
            ++swaps;
        }
        if (pos != -1) {
            if (swaps & 1) sign = -sign;
            int ins = pos - swaps;
            for (int k = n; k > ins; --k) bits[k] = bits[k - 1];
            bits[ins] = b; ++n;
        }
    }
    return sign;
}

constexpr int popcnt4(int v) { int c = 0; for (int i = 0; i < 4; ++i) c += (v >> i) & 1; return c; }

struct Tables {
    float s[16][16];  // sign of e_a * e_b
    float rev[16];    // reverse signs
    float q[16];      // s(i,i)*rev(i): scalar part of R * reverse(R)
};
constexpr Tables make_tables() {
    Tables t{};
    for (int i = 0; i < 16; ++i) {
        int g = popcnt4(i) % 4;
        t.rev[i] = (g == 2 || g == 3) ? -1.0f : 1.0f;
    }
    for (int a = 0; a < 16; ++a)
        for (int b = 0; b < 16; ++b)
            t.s[a][b] = blade_sign(a, b);
    for (int i = 0; i < 16; ++i) t.q[i] = t.s[i][i] * t.rev[i];
    return t;
}
constexpr Tables T = make_tables();

// ---------------------------------------------------------------------------
// Kernel: each block stages 512 elements of R and x into LDS via the CDNA5
// async data mover (coalesced lane-linear b128 loads), then each thread
// computes the sandwich for 2 elements using packed-f32 FMAs.
// ---------------------------------------------------------------------------
#define EPB 512        // elements per block
#define THREADS 256    // 2 elements / thread
#define TILE_BYTES (EPB * 16 * 4)  // 32768 bytes per array

// async b128 load: imm offset applies to BOTH the LDS and global addresses.
#define ALD(ldsoff, vo, sbase, imm)                                           \
    asm volatile("global_load_async_to_lds_b128 %0, %1, %2 offset:" #imm      \
                 :: "v"(ldsoff), "v"(vo), "s"(sbase) : "memory")

__global__ __launch_bounds__(THREADS)
void clifford_sandwich_kernel(const float* __restrict__ gR,
                              const float* __restrict__ gx,
                              float* __restrict__ gout, int ne) {
    __shared__ __align__(16) float smem[2 * EPB * 16];  // 64 KB

    const int t = threadIdx.x;
    const long long blockElem = (long long)blockIdx.x * EPB;
    if (blockElem >= ne) return;

    // ---- async stage R tile into the first 32 KB of LDS, x tile into the second 32 KB ----------
    const uint32_t lds_base = (uint32_t)(uintptr_t)(void*)smem;
    const uint32_t ldsR = lds_base + (uint32_t)t * 16u;
    const uint32_t ldsX = lds_base + (uint32_t)TILE_BYTES + (uint32_t)t * 16u;
    const uint32_t voff = (uint32_t)t * 16u;
    const uint64_t baseR = (uint64_t)(uintptr_t)(gR + blockElem * 16);
    const uint64_t baseX = (uint64_t)(uintptr_t)(gx + blockElem * 16);

    ALD(ldsR, voff, baseR, 0);     ALD(ldsR, voff, baseR, 4096);
    ALD(ldsR, voff, baseR, 8192);  ALD(ldsR, voff, baseR, 12288);
    ALD(ldsR, voff, baseR, 16384); ALD(ldsR, voff, baseR, 20480);
    ALD(ldsR, voff, baseR, 24576); ALD(ldsR, voff, baseR, 28672);
    ALD(ldsX, voff, baseX, 0);     ALD(ldsX, voff, baseX, 4096);
    ALD(ldsX, voff, baseX, 8192);  ALD(ldsX, voff, baseX, 12288);
    ALD(ldsX, voff, baseX, 16384); ALD(ldsX, voff, baseX, 20480);
    ALD(ldsX, voff, baseX, 24576); ALD(ldsX, voff, baseX, 28672);

    asm volatile("s_wait_asynccnt 0" ::: "memory");
    __syncthreads();

    // ---- gather 2 elements/thread from LDS into packed float2 regs ---------
    const int e0 = 2 * t, e1 = 2 * t + 1;
    const float4* s4 = (const float4*)smem;

    v2f R[16], X[16];
#pragma unroll
    for (int k = 0; k < 4; ++k) {
        float4 a = s4[e0 * 4 + k], b = s4[e1 * 4 + k];
        R[k * 4 + 0] = v2f{a.x, b.x}; R[k * 4 + 1] = v2f{a.y, b.y};
        R[k * 4 + 2] = v2f{a.z, b.z}; R[k * 4 + 3] = v2f{a.w, b.w};
        float4 c = s4[EPB * 4 + e0 * 4 + k], d = s4[EPB * 4 + e1 * 4 + k];
        X[k * 4 + 0] = v2f{c.x, d.x}; X[k * 4 + 1] = v2f{c.y, d.y};
        X[k * 4 + 2] = v2f{c.z, d.z}; X[k * 4 + 3] = v2f{c.w, d.w};
    }

    // ---- scalar = <R Rrev>_0 = sum q[i] * R[i]^2 ---------------------------
    v2f sc = v2f{0.0f, 0.0f};
#pragma unroll
    for (int i = 0; i < 16; ++i)
        sc = __builtin_elementwise_fma(T.q[i] < 0.0f ? -R[i] : R[i], R[i], sc);

    // norm_sq = max(|sc|, 1e-8) + 1e-8 ; inv = 1/norm_sq
    const float ns0 = fmaxf(fabsf(sc.x), 1e-8f) + 1e-8f;
    const float ns1 = fmaxf(fabsf(sc.y), 1e-8f) + 1e-8f;
    const v2f inv = v2f{1.0f / ns0, 1.0f / ns1};

    // ---- Rx = gp(R, x): 256 sign-folded packed FMAs ------------------------
    v2f Rx[16];
#pragma unroll
    for (int c = 0; c < 16; ++c) Rx[c] = v2f{0.0f, 0.0f};
#pragma unroll
    for (int a = 0; a < 16; ++a) {
#pragma unroll
        for (int b = 0; b < 16; ++b) {
            Rx[a ^ b] = __builtin_elementwise_fma(
                T.s[a][b] < 0.0f ? -R[a] : R[a], X[b], Rx[a ^ b]);
        }
    }

    // ---- Ri = reverse(R)/norm_sq ------------------------------------------
    v2f Ri[16];
#pragma unroll
    for (int b = 0; b < 16; ++b) {
        v2f v = R[b] * inv;
        Ri[b] = (T.rev[b] < 0.0f) ? -v : v;
    }

    // ---- out = gp(Rx, Ri): 256 sign-folded packed FMAs ---------------------
    v2f O[16];
#pragma unroll
    for (int c = 0; c < 16; ++c) O[c] = v2f{0.0f, 0.0f};
#pragma unroll
    for (int a = 0; a < 16; ++a) {
#pragma unroll
        for (int b = 0; b < 16; ++b) {
            O[a ^ b] = __builtin_elementwise_fma(
                T.s[a][b] < 0.0f ? -Rx[a] : Rx[a], Ri[b], O[a ^ b]);
        }
    }

    // ---- store: 4 x b128 per element --------------------------------------
    if (blockElem + e0 < ne) {
        float4* op = (float4*)(gout + (blockElem + e0) * 16);
#pragma unroll
        for (int k = 0; k < 4; ++k)
            op[k] = float4{O[4 * k + 0].x, O[4 * k + 1].x, O[4 * k + 2].x, O[4 * k + 3].x};
    }
    if (blockElem + e1 < ne) {
        float4* op = (float4*)(gout + (blockElem + e1) * 16);
#pragma unroll
        for (int k = 0; k < 4; ++k)
            op[k] = float4{O[4 * k + 0].y, O[4 * k + 1].y, O[4 * k + 2].y, O[4 * k + 3].y};
    }
}

extern "C" void kernel_launch(void* const* d_in, const int* in_sizes, int n_in,
                              void* d_out, int out_size, void* d_ws, size_t ws_size,
                              hipStream_t stream) {
    const float* R = (const float*)d_in[0];
    const float* x = (const float*)d_in[1];
    // d_in[2] (cayley) and d_in[3] (rev_signs) are baked in at compile time.
    float* out = (float*)d_out;

    const int ne = in_sizes[0] / 16;             // number of multivectors
    const int blocks = (ne + EPB - 1) / EPB;     // 4096 for B=512,S=4096
    clifford_sandwich_kernel<<<dim3(blocks), dim3(THREADS), 0, stream>>>(R, x, out, ne);
}